// DrafterAttention_12506944766511
// MI455X (gfx1250) — compile-verified
//
#include <hip/hip_runtime.h>
#include <hip/hip_bf16.h>
#include <math.h>

// Problem dims (match reference)
#define BB 8
#define HH 16
#define DD 128
#define SS 8192
#define HIDN 2048
#define CHUNK 1024
#define NCHUNK (SS / CHUNK)
#define EPSV 1e-6f

typedef __attribute__((ext_vector_type(2))) float v2f;
typedef __attribute__((ext_vector_type(8))) float v8f;

// -----------------------------------------------------------------------------
// OUT(8x2048) = X(8x2048) . W(2048x2048)^T  i.e. out[m][n] = sum_k X[m][k]*W[n][k]
// One wave per 16-column N-tile; M padded to 16 by clamping rows (pad rows are
// computed but never stored). Full-precision f32 WMMA: V_WMMA_F32_16X16X4_F32.
// A layout (16x4 f32): lane m=l&15; vgpr0 = K (l<16 ? 0 : 2), vgpr1 = K+1.
// B layout (4x16 f32): lane n=l&15; vgpr0 = K (l<16 ? 0 : 2), vgpr1 = K+1.
// C layout (16x16 f32): vgpr r -> row r (lanes 0-15) / row r+8 (lanes 16-31).
// -----------------------------------------------------------------------------
__global__ __launch_bounds__(32) void proj16_wmma(const float* __restrict__ X,
                                                  const float* __restrict__ W,
                                                  float* __restrict__ OUT) {
  const int lane = threadIdx.x;
  const int n0 = blockIdx.x * 16;
  const int mrow = lane & 7;                 // clamp padding rows 8..15 -> dup of 0..7
  const int ksel = (lane < 16) ? 0 : 2;
  const float* xp = X + (size_t)mrow * HIDN + ksel;
  const float* wp = W + (size_t)(n0 + (lane & 15)) * HIDN + ksel;
  v8f c = {0.f, 0.f, 0.f, 0.f, 0.f, 0.f, 0.f, 0.f};
#pragma unroll 8
  for (int k = 0; k < HIDN; k += 4) {
    v2f a = *(const v2f*)(xp + k);
    v2f b = *(const v2f*)(wp + k);
    // 8 args: (neg_a, A, neg_b, B, c_mod, C, reuse_a, reuse_b)
    c = __builtin_amdgcn_wmma_f32_16x16x4_f32(false, a, false, b, (short)0, c,
                                              false, false);
  }
  if (lane < 16) {  // rows 0..7 live in lanes 0..15 (vgpr r -> row r)
    float* o = OUT + n0 + lane;
#pragma unroll
    for (int r = 0; r < 8; ++r) o[(size_t)r * HIDN] = c[r];
  }
}

// -----------------------------------------------------------------------------
// Per-(b,h) RMSNorm over head_dim + RoPE (rotate-half). One block per (b,h).
// -----------------------------------------------------------------------------
__global__ __launch_bounds__(128) void rms_rope(const float* __restrict__ qlin,
                                                const float* __restrict__ gamma,
                                                const float* __restrict__ cosv,
                                                const float* __restrict__ sinv,
                                                float* __restrict__ qout) {
  __shared__ float sred[128];
  __shared__ float sq[128];
  const int bh = blockIdx.x;
  const int d = threadIdx.x;
  float q = qlin[(size_t)bh * DD + d];
  sred[d] = q * q;
  __syncthreads();
#pragma unroll
  for (int off = 64; off > 0; off >>= 1) {
    if (d < off) sred[d] += sred[d + off];
    __syncthreads();
  }
  float rms = rsqrtf(sred[0] * (1.0f / DD) + EPSV);
  float qn = q * rms * gamma[d];
  sq[d] = qn;
  __syncthreads();
  float rot = (d < 64) ? -sq[d + 64] : sq[d - 64];
  qout[(size_t)bh * DD + d] = qn * cosv[d] + rot * sinv[d];
}

// -----------------------------------------------------------------------------
// Flash-decoding partial: grid (bh=128, chunk=8), 256 threads (8 waves).
// Phase A: scores (one coalesced 512B K-row load per wave + butterfly reduce)
// Phase B: chunk softmax numerators in LDS
// Phase C: attn.V (V is s-contiguous: float4 streams, p from LDS as float4)
// Emits per-chunk (m, l, out[128]) for the combine kernel.
// -----------------------------------------------------------------------------
__global__ __launch_bounds__(256) void attn_partial(
    const float* __restrict__ qr, const float* __restrict__ K,
    const float* __restrict__ V, const float* __restrict__ mask,
    float* __restrict__ pout,   // [BH][NCHUNK][DD]
    float* __restrict__ pml) {  // [BH][NCHUNK][2]
  __shared__ __align__(16) float sp[CHUNK];
  __shared__ float sred[8];
  __shared__ float sred2[8];
  const int bh = blockIdx.x;
  const int ck = blockIdx.y;
  const int tid = threadIdx.x;
  const int wv = tid >> 5;
  const int lane = tid & 31;
  const int b = bh >> 4;

  const float4 q4 = *(const float4*)(qr + (size_t)bh * DD + lane * 4);
  const float* Kc = K + ((size_t)bh * SS + (size_t)ck * CHUNK) * DD;
  const float* Vc = V + (size_t)bh * DD * SS + (size_t)ck * CHUNK;
  const float* mk = mask + (size_t)b * SS + (size_t)ck * CHUNK;

  // ---- Phase A: scores
  float wmax = -3.0e38f;
  const int r0 = wv * (CHUNK / 8);
  for (int r = r0; r < r0 + (CHUNK / 8); ++r) {
    float4 k4 = *(const float4*)(Kc + (size_t)r * DD + lane * 4);
    float dp = k4.x * q4.x + k4.y * q4.y + k4.z * q4.z + k4.w * q4.w;
#pragma unroll
    for (int off = 16; off > 0; off >>= 1) dp += __shfl_xor(dp, off, 32);
    dp += mk[r];
    if (lane == 0) sp[r] = dp;
    wmax = fmaxf(wmax, dp);
  }
  if (lane == 0) sred[wv] = wmax;
  __syncthreads();
  float m = sred[0];
#pragma unroll
  for (int i = 1; i < 8; ++i) m = fmaxf(m, sred[i]);

  // ---- Phase B: p = exp(s - m), chunk sum
  float ls = 0.f;
  for (int i = tid; i < CHUNK; i += 256) {
    float p = __expf(sp[i] - m);
    sp[i] = p;
    ls += p;
  }
#pragma unroll
  for (int off = 16; off > 0; off >>= 1) ls += __shfl_xor(ls, off, 32);
  if (lane == 0) sred2[wv] = ls;
  __syncthreads();
  float L = 0.f;
#pragma unroll
  for (int i = 0; i < 8; ++i) L += sred2[i];

  // ---- Phase C: out[d] = sum_s p[s] * V[d][s]
  for (int d = wv; d < DD; d += 8) {
    const float* vrow = Vc + (size_t)d * SS;
    float ax = 0.f, ay = 0.f, az = 0.f, aw = 0.f;
#pragma unroll
    for (int i = 0; i < CHUNK; i += 128) {
      float4 v4 = *(const float4*)(vrow + i + lane * 4);
      float4 p4 = *(const float4*)(&sp[i + lane * 4]);
      ax += p4.x * v4.x;
      ay += p4.y * v4.y;
      az += p4.z * v4.z;
      aw += p4.w * v4.w;
    }
    float t = (ax + ay) + (az + aw);
#pragma unroll
    for (int off = 16; off > 0; off >>= 1) t += __shfl_xor(t, off, 32);
    if (lane == 0) pout[((size_t)bh * NCHUNK + ck) * DD + d] = t;
  }
  if (tid == 0) {
    pml[((size_t)bh * NCHUNK + ck) * 2 + 0] = m;
    pml[((size_t)bh * NCHUNK + ck) * 2 + 1] = L;
  }
}

// -----------------------------------------------------------------------------
// Merge NCHUNK partials per (b,h) -> ctx (B x 2048, head-major, ready for wo)
// -----------------------------------------------------------------------------
__global__ __launch_bounds__(128) void attn_combine(
    const float* __restrict__ pout, const float* __restrict__ pml,
    float* __restrict__ ctx) {
  const int bh = blockIdx.x;
  const int d = threadIdx.x;
  float mv[NCHUNK], lv[NCHUNK];
  float mx = -3.0e38f;
#pragma unroll
  for (int c = 0; c < NCHUNK; ++c) {
    mv[c] = pml[((size_t)bh * NCHUNK + c) * 2 + 0];
    lv[c] = pml[((size_t)bh * NCHUNK + c) * 2 + 1];
    mx = fmaxf(mx, mv[c]);
  }
  float L = 0.f, acc = 0.f;
#pragma unroll
  for (int c = 0; c < NCHUNK; ++c) {
    float w = __expf(mv[c] - mx);
    L += w * lv[c];
    acc += w * pout[((size_t)bh * NCHUNK + c) * DD + d];
  }
  ctx[(size_t)bh * DD + d] = acc / L;
}

// -----------------------------------------------------------------------------
extern "C" void kernel_launch(void* const* d_in, const int* in_sizes, int n_in,
                              void* d_out, int out_size, void* d_ws,
                              size_t ws_size, hipStream_t stream) {
  const float* x    = (const float*)d_in[0];   // (B,1,HID)
  const float* kvk  = (const float*)d_in[1];   // (B,H,S,D)
  const float* kvv  = (const float*)d_in[2];   // (B,H,D,S)
  const float* cosv = (const float*)d_in[3];   // (1,D)
  const float* sinv = (const float*)d_in[4];   // (1,D)
  const float* mask = (const float*)d_in[5];   // (B,1,1,S)
  const float* wq   = (const float*)d_in[6];   // (HID,HID)
  const float* qnw  = (const float*)d_in[7];   // (D,)
  const float* wo   = (const float*)d_in[8];   // (HID,HID)
  float* out = (float*)d_out;                  // (B,1,HID) fp32
  float* ws = (float*)d_ws;

  // workspace layout (floats)
  float* qlin  = ws;                  // 16384
  float* qrope = ws + 16384;          // 16384
  float* pout  = ws + 32768;          // 131072
  float* pml   = ws + 163840;         // 2048
  float* ctx   = ws + 165888;         // 16384   (total ~729 KB)

  proj16_wmma<<<dim3(HIDN / 16), dim3(32), 0, stream>>>(x, wq, qlin);
  rms_rope<<<dim3(BB * HH), dim3(DD), 0, stream>>>(qlin, qnw, cosv, sinv, qrope);
  attn_partial<<<dim3(BB * HH, NCHUNK), dim3(256), 0, stream>>>(qrope, kvk, kvv,
                                                                mask, pout, pml);
  attn_combine<<<dim3(BB * HH), dim3(DD), 0, stream>>>(pout, pml, ctx);
  proj16_wmma<<<dim3(HIDN / 16), dim3(32), 0, stream>>>(ctx, wo, out);
}